// RESGAE_22952305230072
// MI455X (gfx1250) — compile-verified
//
#include <hip/hip_runtime.h>
#include <math.h>

// ---------------------------------------------------------------------------
// RES-GAE forward on gfx1250 (wave32, WMMA).
//   gcn_conv(x,W) = segment_sum((x@W)[src] * w, dst)
// Layers 1-5 computed in reference order (project -> aggregate).
// Layer 6 reordered (aggregate -> project) using linearity of segment_sum:
//   sigmoid(segsum(z1@W6)) == sigmoid(segsum(z1)@W6)
// which shrinks the F=512 aggregation (6.6GB) to an F=32 one (0.4GB).
// GEMMs use V_WMMA_F32_16X16X4_F32; N=50000 = 3125*16 exactly.
// ---------------------------------------------------------------------------

typedef __attribute__((ext_vector_type(2))) float v2f;
typedef __attribute__((ext_vector_type(8))) float v8f;

#define NNODES 50000
#define NEDGES 1600000

// Y[nRows,M] = X[nRows,K] @ W[K,M].  nRows %16==0, M %16==0, K %4==0.
// One wave32 computes one 16x16 tile via v_wmma_f32_16x16x4_f32.
// EPILOGUE: 0 = none, 1 = sigmoid.
template <int EPILOGUE>
__global__ void gemm_wmma_f32(const float* __restrict__ X,
                              const float* __restrict__ W,
                              float* __restrict__ Y,
                              int nRows, int K, int M) {
    const int lane = threadIdx.x & 31;
    const int wid  = (blockIdx.x * blockDim.x + threadIdx.x) >> 5;
    const int colTiles = M >> 4;
    const int rowTile  = wid / colTiles;
    const int colTile  = wid - rowTile * colTiles;
    if (rowTile * 16 >= nRows) return;

    const int half = lane >> 4;   // lanes 16-31 carry K = base+2, base+3
    const int idx  = lane & 15;   // M (A-rows) / N (B-cols) index

    // A: 16x4 f32 tile. lane holds row M=idx; VGPR v holds K = 2*half + v.
    const float* xrow = X + (size_t)(rowTile * 16 + idx) * K + 2 * half;
    // B: 4x16 f32 tile. lane holds col N=idx; VGPR v holds K = 2*half + v.
    const float* wcol = W + colTile * 16 + idx;

    v8f c = {};
    for (int k = 0; k < K; k += 4) {
        v2f a, b;
        a.x = xrow[k];
        a.y = xrow[k + 1];
        b.x = wcol[(size_t)(k + 2 * half) * M];
        b.y = wcol[(size_t)(k + 2 * half + 1) * M];
        // 8 args: (neg_a, A, neg_b, B, c_mod, C, reuse_a, reuse_b)
        c = __builtin_amdgcn_wmma_f32_16x16x4_f32(
                false, a, false, b, (short)0, c, false, false);
    }

    // C/D layout: VGPR r -> row (rowTile*16 + r + 8*half), col (colTile*16+idx)
    float* yb = Y + (size_t)colTile * 16 + idx;
    const int rowBase = rowTile * 16 + 8 * half;
#pragma unroll
    for (int r = 0; r < 8; ++r) {
        float v = c[r];
        if (EPILOGUE == 1) v = 1.0f / (1.0f + expf(-v));
        yb[(size_t)(rowBase + r) * M] = v;
    }
}

// out[dst[e], :] += S[src[e], :] * w[e]   (one thread per edge x 4 features)
__global__ void edge_agg(const float* __restrict__ S,
                         const int* __restrict__ src,
                         const int* __restrict__ dst,
                         const float* __restrict__ w,
                         float* __restrict__ O,
                         int nWork, int Fq /* = F/4 */) {
    int idx = blockIdx.x * blockDim.x + threadIdx.x;
    if (idx >= nWork) return;
    int e  = idx / Fq;
    int c4 = (idx - e * Fq) << 2;
    int F  = Fq << 2;
    const float4 v = *(const float4*)(S + (size_t)src[e] * F + c4);
    const float wt = w[e];
    float* o = O + (size_t)dst[e] * F + c4;
    atomicAdd(o + 0, v.x * wt);
    atomicAdd(o + 1, v.y * wt);
    atomicAdd(o + 2, v.z * wt);
    atomicAdd(o + 3, v.w * wt);
}

__global__ void tanh_ip(float* __restrict__ A, int n) {
    int i = blockIdx.x * blockDim.x + threadIdx.x;
    if (i < n) A[i] = tanhf(A[i]);
}

// A = tanh(A) + R   (residual add)
__global__ void tanh_add_ip(float* __restrict__ A, const float* __restrict__ R,
                            int n) {
    int i = blockIdx.x * blockDim.x + threadIdx.x;
    if (i < n) A[i] = tanhf(A[i]) + R[i];
}

// row-wise softmax over 16 columns
__global__ void softmax16(const float* __restrict__ Z, float* __restrict__ P,
                          int n) {
    int row = blockIdx.x * blockDim.x + threadIdx.x;
    if (row >= n) return;
    const float* z = Z + (size_t)row * 16;
    float m = z[0];
#pragma unroll
    for (int i = 1; i < 16; ++i) m = fmaxf(m, z[i]);
    float e[16];
    float s = 0.0f;
#pragma unroll
    for (int i = 0; i < 16; ++i) { e[i] = expf(z[i] - m); s += e[i]; }
    const float inv = 1.0f / s;
    float* p = P + (size_t)row * 16;
#pragma unroll
    for (int i = 0; i < 16; ++i) p[i] = e[i] * inv;
}

static inline void launch_gemm(const float* X, const float* W, float* Y,
                               int K, int M, int sigmoid, hipStream_t stream) {
    const int waves   = (NNODES / 16) * (M / 16);
    const int threads = waves * 32;
    const int blocks  = (threads + 255) / 256;
    if (sigmoid)
        gemm_wmma_f32<1><<<blocks, 256, 0, stream>>>(X, W, Y, NNODES, K, M);
    else
        gemm_wmma_f32<0><<<blocks, 256, 0, stream>>>(X, W, Y, NNODES, K, M);
}

static inline void launch_agg(const float* S, const int* src, const int* dst,
                              const float* w, float* O, int F,
                              hipStream_t stream) {
    const int nWork  = NEDGES * (F / 4);
    const int blocks = (nWork + 255) / 256;
    edge_agg<<<blocks, 256, 0, stream>>>(S, src, dst, w, O, nWork, F / 4);
}

extern "C" void kernel_launch(void* const* d_in, const int* in_sizes, int n_in,
                              void* d_out, int out_size, void* d_ws,
                              size_t ws_size, hipStream_t stream) {
    const float* feat = (const float*)d_in[0];
    const int*   src  = (const int*)d_in[1];
    const int*   dst  = (const int*)d_in[2];
    const float* ew   = (const float*)d_in[3];
    const float* W1   = (const float*)d_in[4];
    const float* W2   = (const float*)d_in[5];
    const float* W3   = (const float*)d_in[6];
    const float* W4   = (const float*)d_in[7];
    const float* W5   = (const float*)d_in[8];
    const float* W6   = (const float*)d_in[9];

    float* out   = (float*)d_out;
    float* recon = out;                                  // [N,512]
    float* z     = out + (size_t)NNODES * 512;           // [N,16]
    float* pred  = z + (size_t)NNODES * 16;              // [N,16]

    // workspace carve-out (floats)
    float* ws = (float*)d_ws;
    size_t off = 0;
    float* B32a = ws + off; off += (size_t)NNODES * 32;  // S1, then S5
    float* B32b = ws + off; off += (size_t)NNODES * 32;  // x1
    float* B32c = ws + off; off += (size_t)NNODES * 32;  // A5 -> z1
    float* B32d = ws + off; off += (size_t)NNODES * 32;  // G = agg(z1)
    float* B16a = ws + off; off += (size_t)NNODES * 16;  // S2 / S3 / S4
    float* B16b = ws + off; off += (size_t)NNODES * 16;  // x2
    float* B16c = ws + off; off += (size_t)NNODES * 16;  // A4 -> z2

    const size_t b32 = (size_t)NNODES * 32 * sizeof(float);
    const size_t b16 = (size_t)NNODES * 16 * sizeof(float);
    const int n32 = NNODES * 32, n16 = NNODES * 16;
    const int eb32 = (n32 + 255) / 256, eb16 = (n16 + 255) / 256;

    // L1: x1 = tanh(agg(feat @ W1))
    launch_gemm(feat, W1, B32a, 512, 32, 0, stream);
    hipMemsetAsync(B32b, 0, b32, stream);
    launch_agg(B32a, src, dst, ew, B32b, 32, stream);
    tanh_ip<<<eb32, 256, 0, stream>>>(B32b, n32);

    // L2: x2 = tanh(agg(x1 @ W2))
    launch_gemm(B32b, W2, B16a, 32, 16, 0, stream);
    hipMemsetAsync(B16b, 0, b16, stream);
    launch_agg(B16a, src, dst, ew, B16b, 16, stream);
    tanh_ip<<<eb16, 256, 0, stream>>>(B16b, n16);

    // L3: z = tanh(agg(x2 @ W3))   (written straight into d_out)
    launch_gemm(B16b, W3, B16a, 16, 16, 0, stream);
    hipMemsetAsync(z, 0, b16, stream);
    launch_agg(B16a, src, dst, ew, z, 16, stream);
    tanh_ip<<<eb16, 256, 0, stream>>>(z, n16);

    // L4: z2 = tanh(agg(z @ W4)) + x2
    launch_gemm(z, W4, B16a, 16, 16, 0, stream);
    hipMemsetAsync(B16c, 0, b16, stream);
    launch_agg(B16a, src, dst, ew, B16c, 16, stream);
    tanh_add_ip<<<eb16, 256, 0, stream>>>(B16c, B16b, n16);

    // L5: z1 = tanh(agg(z2 @ W5)) + x1
    launch_gemm(B16c, W5, B32a, 16, 32, 0, stream);
    hipMemsetAsync(B32c, 0, b32, stream);
    launch_agg(B32a, src, dst, ew, B32c, 32, stream);
    tanh_add_ip<<<eb32, 256, 0, stream>>>(B32c, B32b, n32);

    // L6 (reordered): recon = sigmoid(agg(z1) @ W6)
    hipMemsetAsync(B32d, 0, b32, stream);
    launch_agg(B32c, src, dst, ew, B32d, 32, stream);
    launch_gemm(B32d, W6, recon, 32, 512, 1, stream);

    // pred = softmax(z)
    softmax16<<<(NNODES + 255) / 256, 256, 0, stream>>>(z, pred, NNODES);
}